// geometric_product_61005715472907
// MI455X (gfx1250) — compile-verified
//
#include <hip/hip_runtime.h>

typedef __attribute__((ext_vector_type(2))) float v2f;
typedef __attribute__((ext_vector_type(4))) float v4f;
typedef __attribute__((ext_vector_type(8))) float v8f;

#define MV 16
#define LDS_STRIDE 260  // 4-float pad: banks (i*260)%64 = 4i -> conflict-free b64 reads, 16B aligned rows

// out[p,i] = sum_{j,k} gp[i,j,k] * x[p,j] * y[p,k]
// GEMM view: out[P,16] = V[P,256] x W[256,16], V[p,jk]=x[p,j]*y[p,k], W[jk,i]=gp[i,j,k]
// One wave32 computes TWO 16x16 tiles (32 points x 16 blades) with 128x V_WMMA_F32_16X16X4_F32,
// sharing each B fragment (gp-derived, tile-invariant) across both tiles: 1 ds_load_b64 -> 2 WMMAs.
__global__ __launch_bounds__(256)
void gp_contract_wmma(const float* __restrict__ gp,
                      const float* __restrict__ x,
                      const float* __restrict__ y,
                      float* __restrict__ out,
                      long P)
{
    __shared__ float gps[MV * LDS_STRIDE];

    const int tid = threadIdx.x;

    // Stage the 16x16x16 Cayley table into padded LDS (16 floats per thread, b128 moves).
    {
        const int i   = tid >> 4;         // blade index i = 0..15
        const int c16 = (tid & 15) * 16;  // 16-float chunk inside the 256-float (j,k) plane
        const v4f* __restrict__ src = (const v4f*)(gp + i * 256 + c16);
        v4f* dst = (v4f*)(&gps[i * LDS_STRIDE + c16]);
        dst[0] = src[0];
        dst[1] = src[1];
        dst[2] = src[2];
        dst[3] = src[3];
    }
    __syncthreads();

    const int  lane = tid & 31;
    const int  row  = lane & 15;  // A row: point-in-tile; B/C column: blade i
    const int  half = lane >> 4;  // selects K parity pair {2h, 2h+1} within each K-group of 4
    const long p0   = ((long)blockIdx.x * 8 + (tid >> 5)) * 32;  // 32 points per wave
    if (p0 + 32 > P) return;      // wave-uniform: EXEC stays all-ones for WMMA

    // x-rows of this lane's two points (j is a compile-time index after unroll).
    const float* __restrict__ xr0 = x + (p0 + row) * MV;
    const float* __restrict__ xr1 = x + (p0 + 16 + row) * MV;
    float xv0[16], xv1[16];
#pragma unroll
    for (int q = 0; q < 4; ++q) {
        v4f t0 = ((const v4f*)xr0)[q];
        v4f t1 = ((const v4f*)xr1)[q];
        xv0[4 * q + 0] = t0.x; xv0[4 * q + 1] = t0.y;
        xv0[4 * q + 2] = t0.z; xv0[4 * q + 3] = t0.w;
        xv1[4 * q + 0] = t1.x; xv1[4 * q + 1] = t1.y;
        xv1[4 * q + 2] = t1.z; xv1[4 * q + 3] = t1.w;
    }

    // y-rows: this lane only ever needs k in {4g+2h, 4g+2h+1}, g=0..3  -> 4x b64 loads per point.
    const float* __restrict__ yr0 = y + (p0 + row) * MV + 2 * half;
    const float* __restrict__ yr1 = y + (p0 + 16 + row) * MV + 2 * half;
    float yp0[8], yp1[8];
#pragma unroll
    for (int g = 0; g < 4; ++g) {
        v2f t0 = *(const v2f*)(yr0 + 4 * g);
        v2f t1 = *(const v2f*)(yr1 + 4 * g);
        yp0[2 * g + 0] = t0.x; yp0[2 * g + 1] = t0.y;
        yp1[2 * g + 0] = t1.x; yp1[2 * g + 1] = t1.y;
    }

    // B-fragment base in LDS: gp[i=row, j, k = 4g + 2h (+1)]
    const float* bbase = &gps[row * LDS_STRIDE + 2 * half];

    v8f acc0 = {0.f, 0.f, 0.f, 0.f, 0.f, 0.f, 0.f, 0.f};
    v8f acc1 = {0.f, 0.f, 0.f, 0.f, 0.f, 0.f, 0.f, 0.f};

#pragma unroll
    for (int t = 0; t < 64; ++t) {
        const int j = t >> 2;  // compile-time after unroll
        const int g = t & 3;   // K-group within j's 16-wide k range
        // B: W[K, i] = gp[i, j, 4g+2h(+1)] -- one conflict-free ds_load_b64, shared by both tiles
        v2f b = *(const v2f*)(bbase + j * 16 + 4 * g);
        // A: V[m, K=4t+2h(+1)] = x[j] * y[4g+2h(+1)]
        v2f a0, a1;
        a0.x = xv0[j] * yp0[2 * g + 0];
        a0.y = xv0[j] * yp0[2 * g + 1];
        a1.x = xv1[j] * yp1[2 * g + 0];
        a1.y = xv1[j] * yp1[2 * g + 1];
        acc0 = __builtin_amdgcn_wmma_f32_16x16x4_f32(
            false, a0, false, b, (short)0, acc0, false, false);
        acc1 = __builtin_amdgcn_wmma_f32_16x16x4_f32(
            false, a1, false, b, (short)0, acc1, false, false);
    }

    // C/D layout: VGPR r -> row (r + 8*half), col = row(lane&15)
    float* __restrict__ orow0 = out + (p0 + 8 * half) * MV + row;
    float* __restrict__ orow1 = out + (p0 + 16 + 8 * half) * MV + row;
#pragma unroll
    for (int r = 0; r < 8; ++r) {
        orow0[(long)r * MV] = acc0[r];
        orow1[(long)r * MV] = acc1[r];
    }
}

extern "C" void kernel_launch(void* const* d_in, const int* in_sizes, int n_in,
                              void* d_out, int out_size, void* d_ws, size_t ws_size,
                              hipStream_t stream) {
    const float* gp = (const float*)d_in[0];  // [16,16,16]
    const float* x  = (const float*)d_in[1];  // [2048,256,16]
    const float* y  = (const float*)d_in[2];  // [2048,256,16]
    float* out = (float*)d_out;               // [2048,256,16]

    const long P     = (long)in_sizes[1] / MV;     // 524288 points
    const long pairs = P / 32;                     // 32 points (2 tiles) per wave
    const int  blocks = (int)((pairs + 7) / 8);    // 8 waves (256 threads) per block

    gp_contract_wmma<<<blocks, 256, 0, stream>>>(gp, x, y, out, P);
}